// SCM_9440338116749
// MI455X (gfx1250) — compile-verified
//
#include <hip/hip_runtime.h>
#include <hip/hip_bf16.h>

typedef __attribute__((ext_vector_type(2))) float v2f;
typedef __attribute__((ext_vector_type(8))) float v8f;

#define PWL_VMIN   (-5.0f)
#define PWL_VMAX   (5.0f)
#define PWL_N      100
#define PWL_INTLEN ((PWL_VMAX - PWL_VMIN) / (float)(PWL_N - 1))   // 10/99

#define DD 16                 // feature dim (fixed by reference)
#define ROWS_PER_BLOCK 128    // 8 waves * 16 rows
#define MAX_BLOCKS 2048       // persistent blocks; grid-stride over tiles

// workspace layout (floats):
//   [0,256)            Winv = inv(I - A), row-major 16x16
//   [256, 256+16*101)  wtab[d][i]  = exp(p[d][i]) + 0.001
//   [256+1616, +1600)  dbtab[d][i] = b[d] + INT_LEN * sum_{j=1..i} wtab[d][j]
#define WS_WINV  0
#define WS_WTAB  256
#define WS_DBTAB (256 + DD * (PWL_N + 1))

// ---------------------------------------------------------------------------
// Prep kernel: one small block builds the inverse and the PWL tables.
// ---------------------------------------------------------------------------
__global__ void scm_prep_kernel(const float* __restrict__ A,
                                const float* __restrict__ p,
                                const float* __restrict__ b,
                                float* __restrict__ ws) {
    __shared__ float M[DD][DD];
    __shared__ float X[DD][DD];

    const int tid = threadIdx.x;

    if (tid < DD) {
        // Per-dim PWL tables: wtab (101) and prefix-sum dbtab (100).
        const int d = tid;
        float* wrow  = ws + WS_WTAB  + d * (PWL_N + 1);
        float* dbrow = ws + WS_DBTAB + d * PWL_N;
        for (int i = 0; i <= PWL_N; ++i) {
            wrow[i] = expf(p[d * (PWL_N + 1) + i]) + 0.001f;
        }
        float acc = b[d];
        dbrow[0] = acc;
        for (int i = 1; i < PWL_N; ++i) {
            acc += PWL_INTLEN * wrow[i];   // delta_h[i-1] = INT_LEN*(exp(p[d][i])+1e-3)
            dbrow[i] = acc;
        }
    } else if (tid == DD) {
        // Gauss-Jordan inverse of (I - A). A = 0.05*randn -> diagonally
        // dominant; no pivoting required.
        for (int r = 0; r < DD; ++r)
            for (int c = 0; c < DD; ++c) {
                M[r][c] = ((r == c) ? 1.0f : 0.0f) - A[r * DD + c];
                X[r][c] = (r == c) ? 1.0f : 0.0f;
            }
        for (int k = 0; k < DD; ++k) {
            const float ip = 1.0f / M[k][k];
            for (int c = 0; c < DD; ++c) { M[k][c] *= ip; X[k][c] *= ip; }
            for (int r = 0; r < DD; ++r) {
                if (r == k) continue;
                const float f = M[r][k];
                for (int c = 0; c < DD; ++c) {
                    M[r][c] -= f * M[k][c];
                    X[r][c] -= f * X[k][c];
                }
            }
        }
        for (int r = 0; r < DD; ++r)
            for (int c = 0; c < DD; ++c)
                ws[WS_WINV + r * DD + c] = X[r][c];
    }
}

// ---------------------------------------------------------------------------
// Main kernel: persistent blocks, grid-stride over 128-row tiles.
// z = eps @ Winv via v_wmma_f32_16x16x4_f32; PWL via LDS tables.
// 256 threads = 8 wave32; each wave owns a 16x16 tile of z.
// ---------------------------------------------------------------------------
__global__ void __launch_bounds__(256)
scm_main_kernel(const float* __restrict__ eps,
                const float* __restrict__ ws,
                float* __restrict__ out,
                int Bn, int numTiles) {
    __shared__ float sW[DD * DD];               // 1 KB
    __shared__ float sWtab[DD * (PWL_N + 1)];   // 6.3 KB
    __shared__ float sDb[DD * PWL_N];           // 6.25 KB
    __shared__ float sEps[ROWS_PER_BLOCK * DD]; // 8 KB

    const int tid = threadIdx.x;

    // Stage the tiny tables (L2-resident) into LDS once per persistent block.
    if (tid < DD * DD) sW[tid] = ws[WS_WINV + tid];
    for (int i = tid; i < DD * (PWL_N + 1); i += 256) sWtab[i] = ws[WS_WTAB + i];
    for (int i = tid; i < DD * PWL_N; i += 256)       sDb[i]   = ws[WS_DBTAB + i];

    const int lane    = tid & 31;
    const int wave    = tid >> 5;
    const int half    = lane >> 4;   // 0: lanes 0-15, 1: lanes 16-31
    const int m       = lane & 15;   // A-row for this lane / D-column index
    const int waveRow = wave * 16;   // row offset of this wave's tile in block
    const float inv_h = 1.0f / PWL_INTLEN;

    for (int tile = blockIdx.x; tile < numTiles; tile += gridDim.x) {
        const long tileBase = (long)tile * ROWS_PER_BLOCK;
        const bool fullTile = (tileBase + ROWS_PER_BLOCK) <= (long)Bn; // uniform

        __syncthreads();   // protect sEps from readers of previous tile

        // Stage this tile's eps with fully coalesced float4 loads.
        const float4* __restrict__ epsv = (const float4*)(eps + tileBase * DD);
        float4* sEpsv = (float4*)sEps;
        if (fullTile) {
#pragma unroll
            for (int i = tid; i < ROWS_PER_BLOCK * DD / 4; i += 256)
                sEpsv[i] = epsv[i];
        } else {
#pragma unroll
            for (int i = tid; i < ROWS_PER_BLOCK * DD / 4; i += 256) {
                const long row = tileBase + (i >> 2);
                float4 v;
                if (row < (long)Bn) { v = epsv[i]; }
                else { v.x = v.y = v.z = v.w = 0.0f; }
                sEpsv[i] = v;
            }
        }
        __syncthreads();

        // D = eps_tile(16x16) x Winv(16x16), K split into 4 chunks of 4.
        // A(16x4 f32): element [m][k] -> lane m+16*(k/2), vgpr k%2.
        // B(4x16 f32): element [k][n] -> vgpr k%2,       lane n+16*(k/2).
        v8f acc = {};
#pragma unroll
        for (int c = 0; c < 4; ++c) {
            v2f a, bf;
            const int kA = 4 * c + 2 * half;          // first K this lane holds
            const float* arow = &sEps[(waveRow + m) * DD + kA];
            a.x = arow[0];
            a.y = arow[1];
            bf.x = sW[(kA + 0) * DD + m];             // n == m == lane&15
            bf.y = sW[(kA + 1) * DD + m];
            acc = __builtin_amdgcn_wmma_f32_16x16x4_f32(
                /*neg_a=*/false, a, /*neg_b=*/false, bf,
                /*c_mod=*/(short)0, acc, /*reuse_a=*/false, /*reuse_b=*/false);
        }

        // PWL: every value in this lane belongs to output dim d = lane&15, so
        // all table gathers hit one 100/101-entry LDS row.
        const int d = m;
        float res[8];
#pragma unroll
        for (int r = 0; r < 8; ++r) {
            const float z = acc[r];
            // searchsorted(points, z, 'right') on the uniform grid
            int idx = (int)floorf((z - PWL_VMIN) * inv_h) + 1;
            idx = idx < 0 ? 0 : (idx > PWL_N ? PWL_N : idx);
            const int spi = idx > 0 ? idx - 1 : 0;
            const float db = sDb[d * PWL_N + spi];
            const float w  = sWtab[d * (PWL_N + 1) + idx];
            const float sp = PWL_VMIN + (float)spi * PWL_INTLEN;
            res[r] = (z - sp) * w + db;
        }

        // D-matrix layout: vgpr r -> row (r + 8*half), lane -> column d.
        float* __restrict__ obase =
            out + (tileBase + waveRow + 8 * half) * DD + d;
        if (fullTile) {
#pragma unroll
            for (int r = 0; r < 8; ++r)
                obase[r * DD] = res[r];       // unguarded: common case
        } else {
#pragma unroll
            for (int r = 0; r < 8; ++r) {
                const long grow = tileBase + waveRow + r + 8 * half;
                if (grow < (long)Bn) out[grow * DD + d] = res[r];
            }
        }
    }
}

// ---------------------------------------------------------------------------
extern "C" void kernel_launch(void* const* d_in, const int* in_sizes, int n_in,
                              void* d_out, int out_size, void* d_ws, size_t ws_size,
                              hipStream_t stream) {
    const float* eps = (const float*)d_in[0];  // [B,16]
    const float* A   = (const float*)d_in[1];  // [16,16]
    const float* p   = (const float*)d_in[2];  // [16,101]
    const float* b   = (const float*)d_in[3];  // [16]
    float* out = (float*)d_out;
    float* ws  = (float*)d_ws;

    const int Bn = in_sizes[0] / DD;
    const int numTiles = (Bn + ROWS_PER_BLOCK - 1) / ROWS_PER_BLOCK;
    const int blocks = numTiles < MAX_BLOCKS ? numTiles : MAX_BLOCKS;

    scm_prep_kernel<<<1, 64, 0, stream>>>(A, p, b, ws);
    scm_main_kernel<<<blocks, 256, 0, stream>>>(eps, ws, out, Bn, numTiles);
}